// MPNNTokenizer_27556510171523
// MI455X (gfx1250) — compile-verified
//
#include <hip/hip_runtime.h>
#include <hip/hip_bf16.h>

#define NN      50000
#define EE      800000
#define NODE_IN 32
#define EDGE_IN 16
#define HID     128
#define OUTD    128
#define NLAYERS 2
#define KMSG    (2*HID + EDGE_IN)   // 272
#define KUPD    (2*HID)             // 256

typedef __attribute__((ext_vector_type(2))) float v2f;
typedef __attribute__((ext_vector_type(8))) float v8f;

// ---------------------------------------------------------------------------
// WMMA helpers: V_WMMA_F32_16X16X4_F32,  D(16x16,f32) = A(16x4,f32) x B(4x16,f32) + C
// A frag (2 VGPRs): lane l: m=l&15, kh=(l>>4)*2 -> {A[m][k0+kh], A[m][k0+kh+1]}
// B frag (2 VGPRs): lane l: n=l&15, kh=(l>>4)*2 -> {B[k0+kh][n], B[k0+kh+1][n]}
// C/D (8 VGPRs):    lane l: n=l&15, vgpr v holds m = v + 8*(l>>4)
// ---------------------------------------------------------------------------
__device__ __forceinline__ v8f wmma_f32(v2f a, v2f b, v8f c) {
  return __builtin_amdgcn_wmma_f32_16x16x4_f32(false, a, false, b, (short)0, c,
                                               false, false);
}

__device__ __forceinline__ v2f frag_a(const float* As, int lda, int m, int kh, int k0) {
  v2f a;
  a.x = As[m * lda + k0 + kh];
  a.y = As[m * lda + k0 + kh + 1];
  return a;
}
__device__ __forceinline__ v2f frag_b(const float* Bs, int ldb, int n, int kh, int k0) {
  v2f b;
  b.x = Bs[(k0 + kh) * ldb + n];
  b.y = Bs[(k0 + kh + 1) * ldb + n];
  return b;
}

__device__ __forceinline__ float reluf(float x) { return x > 0.f ? x : 0.f; }

// ---------------------------------------------------------------------------
// degree / inverse-denominator
// ---------------------------------------------------------------------------
__global__ __launch_bounds__(256) void degree_kernel(const int* __restrict__ eidx,
                                                     float* __restrict__ cnt, int E) {
  int e = blockIdx.x * 256 + threadIdx.x;
  if (e < E) unsafeAtomicAdd(&cnt[eidx[E + e]], 1.0f);   // tgt = edge_index[1]
}

__global__ __launch_bounds__(256) void invden_kernel(float* __restrict__ cnt, int n) {
  int i = blockIdx.x * 256 + threadIdx.x;
  if (i < n) cnt[i] = 1.0f / fmaxf(cnt[i], 1.0f);
}

// ---------------------------------------------------------------------------
// Encoder: h = LN(relu(x @ enc_W + enc_b)).  16-node tile, 128 thr = 4 waves.
// ---------------------------------------------------------------------------
__global__ __launch_bounds__(128)
void encoder_kernel(const float* __restrict__ x, const float* __restrict__ W,
                    const float* __restrict__ b, const float* __restrict__ g,
                    const float* __restrict__ beta, float* __restrict__ h) {
  __shared__ float As[16 * 33];
  __shared__ float Ws[32 * 136];
  __shared__ float Ts[16 * 136];
  __shared__ float red[2 * 16 * 8];
  __shared__ float s_mu[16], s_rs[16];

  const int tid = threadIdx.x, lane = tid & 31, wave = tid >> 5;
  const int node0 = blockIdx.x * 16;
  const int m16 = lane & 15, kh = (lane >> 4) << 1;

  for (int i = tid; i < 16 * NODE_IN; i += 128) {
    int row = i >> 5, col = i & 31;
    As[row * 33 + col] = x[(node0 + row) * NODE_IN + col];
  }
  for (int i = tid; i < NODE_IN * HID; i += 128) {
    int kk = i >> 7, n = i & 127;
    Ws[kk * 136 + n] = W[kk * HID + n];
  }
  __syncthreads();

  const v8f vzero = {0.f, 0.f, 0.f, 0.f, 0.f, 0.f, 0.f, 0.f};
  v8f acc[2] = {vzero, vzero};
#pragma unroll
  for (int kk = 0; kk < NODE_IN; kk += 4) {
    v2f a = frag_a(As, 33, m16, kh, kk);
#pragma unroll
    for (int nt = 0; nt < 2; ++nt) {
      v2f bf = frag_b(Ws, 136, wave * 32 + nt * 16 + m16, kh, kk);
      acc[nt] = wmma_f32(a, bf, acc[nt]);
    }
  }
#pragma unroll
  for (int nt = 0; nt < 2; ++nt) {
    int n = wave * 32 + nt * 16 + m16;
    float bias = b[n];
#pragma unroll
    for (int v = 0; v < 8; ++v) {
      int m = v + ((lane >> 4) << 3);
      Ts[m * 136 + n] = reluf(acc[nt][v] + bias);
    }
  }
  __syncthreads();

  // LayerNorm over 128 cols: 8 threads per row
  int row = tid >> 3, sub = tid & 7;
  float s = 0.f, ss = 0.f;
  for (int c = sub * 16; c < sub * 16 + 16; ++c) {
    float t = Ts[row * 136 + c];
    s += t; ss += t * t;
  }
  red[row * 8 + sub] = s;
  red[128 + row * 8 + sub] = ss;
  __syncthreads();
  if (sub == 0) {
    float S = 0.f, SS = 0.f;
    for (int j = 0; j < 8; ++j) { S += red[row * 8 + j]; SS += red[128 + row * 8 + j]; }
    float mu = S * (1.0f / HID);
    float var = SS * (1.0f / HID) - mu * mu;
    s_mu[row] = mu;
    s_rs[row] = rsqrtf(var + 1e-5f);
  }
  __syncthreads();
  float mu = s_mu[row], rs = s_rs[row];
  for (int c = sub * 16; c < sub * 16 + 16; ++c) {
    float t = (Ts[row * 136 + c] - mu) * rs;
    h[(node0 + row) * HID + c] = t * g[c] + beta[c];
  }
}

// ---------------------------------------------------------------------------
// Edge message MLP (fused 2 GEMMs + scatter-add):
//   m = relu([h_tgt|h_src|edge_attr] @ W1 + b1) @ W2 + b2 ; aggr[tgt] += m
// 64-edge tile, 256 thr = 8 waves; wave w: m-tile = w>>1, 64-col half = (w&1).
// ---------------------------------------------------------------------------
__global__ __launch_bounds__(256)
void edge_msg_kernel(const float* __restrict__ h, const float* __restrict__ edge_attr,
                     const int* __restrict__ eidx, const float* __restrict__ W1,
                     const float* __restrict__ b1, const float* __restrict__ W2,
                     const float* __restrict__ b2, float* __restrict__ aggr, int E) {
  __shared__ float As[64 * 33];
  __shared__ float Ws[32 * 136];
  __shared__ float Ts[64 * 136];
  __shared__ int s_src[64], s_tgt[64];

  const int tid = threadIdx.x, lane = tid & 31, wave = tid >> 5;
  const int mt = wave >> 1;
  const int colbase = (wave & 1) * 64;
  const int e0 = blockIdx.x * 64;
  const int m16 = lane & 15, kh = (lane >> 4) << 1;
  const int mrow = mt * 16 + m16;

  if (tid < 64) {
    s_src[tid] = eidx[e0 + tid];       // edge_index[0] = src
    s_tgt[tid] = eidx[E + e0 + tid];   // edge_index[1] = tgt
  }
  __syncthreads();

  const v8f vzero = {0.f, 0.f, 0.f, 0.f, 0.f, 0.f, 0.f, 0.f};
  v8f acc[4] = {vzero, vzero, vzero, vzero};

  // ---- GEMM1: [64 x 272] @ [272 x 128], K chunks of 32 (last = 16) ----
  for (int kc = 0; kc < KMSG; kc += 32) {
    const int ck = (KMSG - kc) >= 32 ? 32 : 16;
    for (int i = tid; i < 64 * ck; i += 256) {
      int rrow, col;
      if (ck == 32) { rrow = i >> 5; col = i & 31; }
      else          { rrow = i >> 4; col = i & 15; }
      int gcol = kc + col;
      float v;
      if (gcol < HID)            v = h[s_tgt[rrow] * HID + gcol];
      else if (gcol < 2 * HID)   v = h[s_src[rrow] * HID + (gcol - HID)];
      else                       v = edge_attr[(e0 + rrow) * EDGE_IN + (gcol - 2 * HID)];
      As[rrow * 33 + col] = v;
    }
    for (int i = tid; i < ck * HID; i += 256) {
      int kk = i >> 7, n = i & 127;
      Ws[kk * 136 + n] = W1[(kc + kk) * HID + n];
    }
    __syncthreads();
    for (int kk = 0; kk < ck; kk += 4) {
      v2f a = frag_a(As, 33, mrow, kh, kk);
#pragma unroll
      for (int nt = 0; nt < 4; ++nt) {
        v2f bf = frag_b(Ws, 136, colbase + nt * 16 + m16, kh, kk);
        acc[nt] = wmma_f32(a, bf, acc[nt]);
      }
    }
    __syncthreads();
  }

  // relu + b1 -> Ts[64][128]
#pragma unroll
  for (int nt = 0; nt < 4; ++nt) {
    int n = colbase + nt * 16 + m16;
    float bias = b1[n];
#pragma unroll
    for (int v = 0; v < 8; ++v) {
      int m = mt * 16 + v + ((lane >> 4) << 3);
      Ts[m * 136 + n] = reluf(acc[nt][v] + bias);
    }
  }
  __syncthreads();

  // ---- GEMM2: Ts[64 x 128] @ W2[128 x 128] ----
  v8f acc2[4] = {vzero, vzero, vzero, vzero};
  for (int kc = 0; kc < HID; kc += 32) {
    for (int i = tid; i < 32 * HID; i += 256) {
      int kk = i >> 7, n = i & 127;
      Ws[kk * 136 + n] = W2[(kc + kk) * HID + n];
    }
    __syncthreads();
    for (int kk = 0; kk < 32; kk += 4) {
      v2f a = frag_a(Ts, 136, mrow, kh, kc + kk);
#pragma unroll
      for (int nt = 0; nt < 4; ++nt) {
        v2f bf = frag_b(Ws, 136, colbase + nt * 16 + m16, kh, kk);
        acc2[nt] = wmma_f32(a, bf, acc2[nt]);
      }
    }
    __syncthreads();
  }

  // bias + fused scatter-add (message tensor never hits memory)
#pragma unroll
  for (int nt = 0; nt < 4; ++nt) {
    int n = colbase + nt * 16 + m16;
    float bias = b2[n];
#pragma unroll
    for (int v = 0; v < 8; ++v) {
      int m = mt * 16 + v + ((lane >> 4) << 3);
      unsafeAtomicAdd(&aggr[s_tgt[m] * HID + n], acc2[nt][v] + bias);
    }
  }
}

// ---------------------------------------------------------------------------
// Node update: h = LN(h + relu([h | aggr*invden] @ W1 + b1) @ W2 + b2)
// 16-node tile, 128 thr = 4 waves.
// ---------------------------------------------------------------------------
__global__ __launch_bounds__(128)
void node_update_kernel(float* __restrict__ h, const float* __restrict__ aggr,
                        const float* __restrict__ invden, const float* __restrict__ W1,
                        const float* __restrict__ b1, const float* __restrict__ W2,
                        const float* __restrict__ b2, const float* __restrict__ g,
                        const float* __restrict__ beta) {
  __shared__ float As[16 * 33];
  __shared__ float Ws[32 * 136];
  __shared__ float Ts[16 * 136];
  __shared__ float red[2 * 16 * 8];
  __shared__ float s_mu[16], s_rs[16];

  const int tid = threadIdx.x, lane = tid & 31, wave = tid >> 5;
  const int node0 = blockIdx.x * 16;
  const int m16 = lane & 15, kh = (lane >> 4) << 1;

  const v8f vzero = {0.f, 0.f, 0.f, 0.f, 0.f, 0.f, 0.f, 0.f};
  v8f acc[2] = {vzero, vzero};

  // ---- GEMM1: [16 x 256] @ [256 x 128] ----
  for (int kc = 0; kc < KUPD; kc += 32) {
    for (int i = tid; i < 16 * 32; i += 128) {
      int row = i >> 5, col = i & 31;
      int gcol = kc + col, node = node0 + row;
      float v = (gcol < HID) ? h[node * HID + gcol]
                             : aggr[node * HID + (gcol - HID)] * invden[node];
      As[row * 33 + col] = v;
    }
    for (int i = tid; i < 32 * HID; i += 128) {
      int kk = i >> 7, n = i & 127;
      Ws[kk * 136 + n] = W1[(kc + kk) * HID + n];
    }
    __syncthreads();
    for (int kk = 0; kk < 32; kk += 4) {
      v2f a = frag_a(As, 33, m16, kh, kk);
#pragma unroll
      for (int nt = 0; nt < 2; ++nt) {
        v2f bf = frag_b(Ws, 136, wave * 32 + nt * 16 + m16, kh, kk);
        acc[nt] = wmma_f32(a, bf, acc[nt]);
      }
    }
    __syncthreads();
  }

#pragma unroll
  for (int nt = 0; nt < 2; ++nt) {
    int n = wave * 32 + nt * 16 + m16;
    float bias = b1[n];
#pragma unroll
    for (int v = 0; v < 8; ++v) {
      int m = v + ((lane >> 4) << 3);
      Ts[m * 136 + n] = reluf(acc[nt][v] + bias);
    }
  }
  __syncthreads();

  // ---- GEMM2: Ts[16 x 128] @ W2[128 x 128] ----
  v8f acc2[2] = {vzero, vzero};
  for (int kc = 0; kc < HID; kc += 32) {
    for (int i = tid; i < 32 * HID; i += 128) {
      int kk = i >> 7, n = i & 127;
      Ws[kk * 136 + n] = W2[(kc + kk) * HID + n];
    }
    __syncthreads();
    for (int kk = 0; kk < 32; kk += 4) {
      v2f a = frag_a(Ts, 136, m16, kh, kc + kk);
#pragma unroll
      for (int nt = 0; nt < 2; ++nt) {
        v2f bf = frag_b(Ws, 136, wave * 32 + nt * 16 + m16, kh, kk);
        acc2[nt] = wmma_f32(a, bf, acc2[nt]);
      }
    }
    __syncthreads();
  }

  // residual -> Ts
#pragma unroll
  for (int nt = 0; nt < 2; ++nt) {
    int n = wave * 32 + nt * 16 + m16;
    float bias = b2[n];
#pragma unroll
    for (int v = 0; v < 8; ++v) {
      int m = v + ((lane >> 4) << 3);
      Ts[m * 136 + n] = h[(node0 + m) * HID + n] + acc2[nt][v] + bias;
    }
  }
  __syncthreads();

  // LayerNorm + in-place write
  int row = tid >> 3, sub = tid & 7;
  float s = 0.f, ss = 0.f;
  for (int c = sub * 16; c < sub * 16 + 16; ++c) {
    float t = Ts[row * 136 + c];
    s += t; ss += t * t;
  }
  red[row * 8 + sub] = s;
  red[128 + row * 8 + sub] = ss;
  __syncthreads();
  if (sub == 0) {
    float S = 0.f, SS = 0.f;
    for (int j = 0; j < 8; ++j) { S += red[row * 8 + j]; SS += red[128 + row * 8 + j]; }
    float mu = S * (1.0f / HID);
    float var = SS * (1.0f / HID) - mu * mu;
    s_mu[row] = mu;
    s_rs[row] = rsqrtf(var + 1e-5f);
  }
  __syncthreads();
  float mu = s_mu[row], rs = s_rs[row];
  for (int c = sub * 16; c < sub * 16 + 16; ++c) {
    float t = (Ts[row * 136 + c] - mu) * rs;
    h[(node0 + row) * HID + c] = t * g[c] + beta[c];
  }
}

// ---------------------------------------------------------------------------
// Head: out = h @ head_W + head_b.  16-node tile, 128 thr = 4 waves.
// ---------------------------------------------------------------------------
__global__ __launch_bounds__(128)
void head_kernel(const float* __restrict__ h, const float* __restrict__ W,
                 const float* __restrict__ b, float* __restrict__ out) {
  __shared__ float As[16 * 33];
  __shared__ float Ws[32 * 136];

  const int tid = threadIdx.x, lane = tid & 31, wave = tid >> 5;
  const int node0 = blockIdx.x * 16;
  const int m16 = lane & 15, kh = (lane >> 4) << 1;

  const v8f vzero = {0.f, 0.f, 0.f, 0.f, 0.f, 0.f, 0.f, 0.f};
  v8f acc[2] = {vzero, vzero};

  for (int kc = 0; kc < HID; kc += 32) {
    for (int i = tid; i < 16 * 32; i += 128) {
      int row = i >> 5, col = i & 31;
      As[row * 33 + col] = h[(node0 + row) * HID + kc + col];
    }
    for (int i = tid; i < 32 * OUTD; i += 128) {
      int kk = i >> 7, n = i & 127;
      Ws[kk * 136 + n] = W[(kc + kk) * OUTD + n];
    }
    __syncthreads();
    for (int kk = 0; kk < 32; kk += 4) {
      v2f a = frag_a(As, 33, m16, kh, kk);
#pragma unroll
      for (int nt = 0; nt < 2; ++nt) {
        v2f bf = frag_b(Ws, 136, wave * 32 + nt * 16 + m16, kh, kk);
        acc[nt] = wmma_f32(a, bf, acc[nt]);
      }
    }
    __syncthreads();
  }
#pragma unroll
  for (int nt = 0; nt < 2; ++nt) {
    int n = wave * 32 + nt * 16 + m16;
    float bias = b[n];
#pragma unroll
    for (int v = 0; v < 8; ++v) {
      int m = v + ((lane >> 4) << 3);
      out[(node0 + m) * OUTD + n] = acc[nt][v] + bias;
    }
  }
}

// ---------------------------------------------------------------------------
extern "C" void kernel_launch(void* const* d_in, const int* in_sizes, int n_in,
                              void* d_out, int out_size, void* d_ws, size_t ws_size,
                              hipStream_t stream) {
  const float* x         = (const float*)d_in[0];
  const int*   eidx      = (const int*)d_in[1];      // [2][E] int32
  const float* edge_attr = (const float*)d_in[2];
  const float* enc_W     = (const float*)d_in[3];
  const float* enc_b     = (const float*)d_in[4];
  const float* enc_g     = (const float*)d_in[5];
  const float* enc_beta  = (const float*)d_in[6];
  const float* msg_W1    = (const float*)d_in[7];    // [L][272][128]
  const float* msg_b1    = (const float*)d_in[8];
  const float* msg_W2    = (const float*)d_in[9];    // [L][128][128]
  const float* msg_b2    = (const float*)d_in[10];
  const float* upd_W1    = (const float*)d_in[11];   // [L][256][128]
  const float* upd_b1    = (const float*)d_in[12];
  const float* upd_W2    = (const float*)d_in[13];   // [L][128][128]
  const float* upd_b2    = (const float*)d_in[14];
  const float* ln_g      = (const float*)d_in[15];
  const float* ln_beta   = (const float*)d_in[16];
  const float* head_W    = (const float*)d_in[17];
  const float* head_b    = (const float*)d_in[18];

  float* ws   = (float*)d_ws;
  float* h    = ws;                       // N*128 floats (25.6 MB, L2-resident)
  float* aggr = h + (size_t)NN * HID;     // N*128 floats
  float* den  = aggr + (size_t)NN * HID;  // N floats (cnt -> invden)

  // degree -> 1/max(cnt,1)
  hipMemsetAsync(den, 0, NN * sizeof(float), stream);
  degree_kernel<<<(EE + 255) / 256, 256, 0, stream>>>(eidx, den, EE);
  invden_kernel<<<(NN + 255) / 256, 256, 0, stream>>>(den, NN);

  // encoder
  encoder_kernel<<<NN / 16, 128, 0, stream>>>(x, enc_W, enc_b, enc_g, enc_beta, h);

  for (int l = 0; l < NLAYERS; ++l) {
    hipMemsetAsync(aggr, 0, (size_t)NN * HID * sizeof(float), stream);
    edge_msg_kernel<<<EE / 64, 256, 0, stream>>>(
        h, edge_attr, eidx,
        msg_W1 + (size_t)l * KMSG * HID, msg_b1 + (size_t)l * HID,
        msg_W2 + (size_t)l * HID * HID, msg_b2 + (size_t)l * HID, aggr, EE);
    node_update_kernel<<<NN / 16, 128, 0, stream>>>(
        h, aggr, den,
        upd_W1 + (size_t)l * KUPD * HID, upd_b1 + (size_t)l * HID,
        upd_W2 + (size_t)l * HID * HID, upd_b2 + (size_t)l * HID,
        ln_g + (size_t)l * HID, ln_beta + (size_t)l * HID);
  }

  head_kernel<<<NN / 16, 128, 0, stream>>>(h, head_W, head_b, (float*)d_out);
}